// CBDiceLoss_83648783057223
// MI455X (gfx1250) — compile-verified
//
#include <hip/hip_runtime.h>

// Flip to 0 if the gfx1250 assembler ever rejects the async-to-LDS mnemonics.
#define USE_ASYNC_LDS 1

constexpr int   H          = 768;
constexpr int   W          = 768;
constexpr int   NIMG       = 32;
constexpr int   T          = 32;          // interior tile edge
constexpr int   RHALO      = 16;          // halo radius (>= max erosion depth)
constexpr int   PAD        = T + 2*RHALO; // 64
constexpr int   NTHREADS   = 256;         // 8 waves (wave32)
constexpr int   SKEL_ITERS = 10;
constexpr int   RAD_ITERS  = 16;
constexpr float EPS        = 1e-6f;

// ---------------------------------------------------------------------------
__global__ void cbdice_zero_ws(float* __restrict__ w, int n) {
  int i = blockIdx.x * blockDim.x + threadIdx.x;
  if (i < n) w[i] = 0.0f;
}

__device__ __forceinline__ float hmin3(const float* r) {
  return fminf(fminf(r[-1], r[0]), r[1]);
}
__device__ __forceinline__ float hmax3(const float* r) {
  return fmaxf(fmaxf(r[-1], r[0]), r[1]);
}

// Separable 3x3 erode cur->nxt restricted to [k,PAD-k)^2.
// Thread owns column `ecol`, rows [er0, er0+16). Rolling row-min registers:
// ~3 DS loads + 4 VALU per produced cell, lane->column stride-1 (bank clean).
__device__ __forceinline__ void erode_pass(const float* __restrict__ cur,
                                           float* __restrict__ nxt,
                                           int k, int ecol, int er0) {
  const int lo = k, hi = PAD - k;
  if (ecol < lo || ecol >= hi) return;
  int rs = er0     > lo ? er0      : lo;
  int re = er0 + 16 < hi ? er0 + 16 : hi;
  if (rs >= re) return;
  const float* p = cur + ecol;
  float a = hmin3(p + (rs - 1) * PAD);
  float b = hmin3(p + rs * PAD);
  for (int r = rs; r < re; ++r) {
    float c = hmin3(p + (r + 1) * PAD);
    nxt[r * PAD + ecol] = fminf(fminf(a, b), c);
    a = b; b = c;
  }
}

// ---------------------------------------------------------------------------
// Fused per-tile kernel. LDS: only the two erosion ping-pong buffers (33 KB).
// All pointwise state (skel, radius acc, x, saved G results) lives in
// unrolled register arrays owned by a fixed thread<->cell mapping.
// ---------------------------------------------------------------------------
__global__ __launch_bounds__(NTHREADS)
void cbdice_tile_kernel(const float* __restrict__ P,
                        const float* __restrict__ G,
                        float* __restrict__ sums /* NIMG*4 */) {
  __shared__ float eA[PAD * PAD];   // 16 KB erosion ping
  __shared__ float eB[PAD * PAD];   // 16 KB erosion pong
  __shared__ float red[8 * 4];      // cross-wave reduction

  const int tid  = threadIdx.x;
  const int tx   = blockIdx.x;
  const int ty   = blockIdx.y;
  const int img  = blockIdx.z;
  const int gx0  = tx * T - RHALO;
  const int gy0  = ty * T - RHALO;
  const bool edge = (tx == 0) || (ty == 0) || (tx == W / T - 1) || (ty == H / T - 1);

  // erode mapping: 64 columns x 4 strips of 16 rows
  const int ecol = tid & 63;
  const int er0  = (tid >> 6) * 16;
  // interior ownership mapping: 32 columns x 8 strips of 4 rows
  const int icol = tid & 31;            // ix
  const int ir0  = (tid >> 5) * 4;      // iy base

  // saved G-phase results (registers)
  float gskl[4], grr[4], gxx[4];
  float a = 0.f, b = 0.f, c = 0.f, d = 0.f;

  for (int phase = 0; phase < 2; ++phase) {
    const float* X    = (phase == 0) ? G : P;
    const float* base = X + (size_t)img * ((size_t)H * W);

    // ---- stage padded tile into LDS (+inf outside image), async B128 -----
    #pragma unroll
    for (int j = 0; j < (PAD * PAD / 4) / NTHREADS; ++j) {   // 4
      int cell = tid + j * NTHREADS;           // 0..1023 float4 cells
      int py   = cell >> 4;                    // 16 float4 per padded row
      int qx   = cell & 15;
      int gy   = gy0 + py;
      int gxf  = gx0 + qx * 4;
      float* ldst = &eA[py * PAD + qx * 4];
      bool in = (gy >= 0) && (gy < H) && (gxf >= 0) && (gxf + 3 < W);
      if (in) {
#if USE_ASYNC_LDS
        unsigned     loff = (unsigned)(size_t)ldst;          // LDS byte addr
        const float* gsrc = base + (size_t)gy * W + gxf;
        asm volatile("global_load_async_to_lds_b128 %0, %1, off"
                     :: "v"(loff), "v"(gsrc) : "memory");
#else
        *(float4*)ldst = *(const float4*)(base + (size_t)gy * W + gxf);
#endif
      } else {
        ldst[0] = __builtin_inff(); ldst[1] = __builtin_inff();
        ldst[2] = __builtin_inff(); ldst[3] = __builtin_inff();
      }
    }
#if USE_ASYNC_LDS
    asm volatile("s_wait_asynccnt 0" ::: "memory");
#endif
    __syncthreads();

    // ---- capture interior x (registers), init pointwise state ------------
    float xin[4], accv[4], skl[4];
    #pragma unroll
    for (int j = 0; j < 4; ++j) {
      xin[j]  = eA[(ir0 + j + RHALO) * PAD + (icol + RHALO)];
      accv[j] = 0.f;
      skl[j]  = 0.f;
    }
    __syncthreads();   // all of eA staged & sampled before erosion overwrites

    float* cur = eA;   // e_{k-1}
    float* nxt = eB;   // e_k

    // ---- passes 1..11: erode + fused dilate/skeleton/radius --------------
    for (int k = 1; k <= SKEL_ITERS + 1; ++k) {
      erode_pass(cur, nxt, k, ecol, er0);
      __syncthreads();

      if (!edge) {
        // separable dilate over owned 4-row interior strip
        float hm[6], cen[6];
        #pragma unroll
        for (int r = 0; r < 6; ++r) {
          const float* row = nxt + (ir0 + 15 + r) * PAD + (icol + RHALO);
          cen[r] = row[0];
          hm[r]  = fmaxf(fmaxf(row[-1], cen[r]), row[1]);
        }
        #pragma unroll
        for (int j = 0; j < 4; ++j) {
          float mx = fmaxf(fmaxf(hm[j], hm[j + 1]), hm[j + 2]);
          accv[j] += cen[j + 1];                      // acc += e_k
          float ep = cur[(ir0 + j + RHALO) * PAD + (icol + RHALO)];
          float delta = fmaxf(ep - mx, 0.f);
          if (k == 1) skl[j] = delta;
          else {
            float s = skl[j] + (1.f - skl[j]) * delta;
            skl[j] = fminf(fmaxf(s, 0.f), 1.f);
          }
        }
      } else {
        // boundary tile: coordinate-checked 3x3 max (OOB acts as -inf)
        #pragma unroll
        for (int j = 0; j < 4; ++j) {
          int iy = ir0 + j, ix = icol;
          int py = iy + RHALO, px = ix + RHALO;
          int gy = ty * T + iy, gx = tx * T + ix;
          float ekc = nxt[py * PAD + px];
          accv[j] += ekc;
          float mx = ekc;   // center always in-image
          #pragma unroll
          for (int dy = -1; dy <= 1; ++dy) {
            int yy = gy + dy;
            if (yy < 0 || yy >= H) continue;
            #pragma unroll
            for (int dx = -1; dx <= 1; ++dx) {
              int xx = gx + dx;
              if (xx < 0 || xx >= W) continue;
              mx = fmaxf(mx, nxt[(py + dy) * PAD + (px + dx)]);
            }
          }
          float delta = fmaxf(cur[py * PAD + px] - mx, 0.f);
          if (k == 1) skl[j] = delta;
          else {
            float s = skl[j] + (1.f - skl[j]) * delta;
            skl[j] = fminf(fmaxf(s, 0.f), 1.f);
          }
        }
      }
      __syncthreads();
      float* t = cur; cur = nxt; nxt = t;
    }

    // ---- passes 12..16: erode + radius accumulation only -----------------
    for (int k = SKEL_ITERS + 2; k <= RAD_ITERS; ++k) {
      erode_pass(cur, nxt, k, ecol, er0);
      __syncthreads();
      #pragma unroll
      for (int j = 0; j < 4; ++j)
        accv[j] += nxt[(ir0 + j + RHALO) * PAD + (icol + RHALO)];
      // no second barrier: next pass only reads nxt, writes retired cur
      float* t = cur; cur = nxt; nxt = t;
    }

    // ---- radius map (fold +eps), all in registers ------------------------
    #pragma unroll
    for (int j = 0; j < 4; ++j) {
      float rmap = fminf(fmaxf(accv[j] * (1.f / RAD_ITERS), 0.f), 1.f);
      accv[j] = rmap * xin[j] + EPS;        // r(+eps)
    }

    if (phase == 0) {
      #pragma unroll
      for (int j = 0; j < 4; ++j) { gskl[j] = skl[j]; grr[j] = accv[j]; gxx[j] = xin[j]; }
    } else {
      #pragma unroll
      for (int j = 0; j < 4; ++j) {
        a += skl[j]  * gxx[j] * grr[j];     // Pskel * G * rG
        b += skl[j]  * grr[j];              // Pskel * rG
        c += gskl[j] * xin[j] * accv[j];    // Gskel * P * rP
        d += gskl[j] * accv[j];             // Gskel * rP
      }
    }
    __syncthreads();   // everyone done reading e-buffers before next staging
  }

  // ---- block reduction: wave32 shuffles, then cross-wave via LDS ---------
  for (int off = 16; off > 0; off >>= 1) {
    a += __shfl_down(a, off);
    b += __shfl_down(b, off);
    c += __shfl_down(c, off);
    d += __shfl_down(d, off);
  }
  int wave = tid >> 5, lane = tid & 31;
  if (lane == 0) {
    red[wave * 4 + 0] = a; red[wave * 4 + 1] = b;
    red[wave * 4 + 2] = c; red[wave * 4 + 3] = d;
  }
  __syncthreads();
  if (tid == 0) {
    float A = 0.f, B = 0.f, C = 0.f, D = 0.f;
    #pragma unroll
    for (int w2 = 0; w2 < NTHREADS / 32; ++w2) {
      A += red[w2 * 4 + 0]; B += red[w2 * 4 + 1];
      C += red[w2 * 4 + 2]; D += red[w2 * 4 + 3];
    }
    atomicAdd(&sums[img * 4 + 0], A);
    atomicAdd(&sums[img * 4 + 1], B);
    atomicAdd(&sums[img * 4 + 2], C);
    atomicAdd(&sums[img * 4 + 3], D);
  }
}

// ---------------------------------------------------------------------------
__global__ void cbdice_finalize(const float* __restrict__ sums,
                                float* __restrict__ out) {
  int n = threadIdx.x;  // one wave of 32 lanes == NIMG
  float A = sums[n * 4 + 0], B = sums[n * 4 + 1];
  float C = sums[n * 4 + 2], D = sums[n * 4 + 3];
  float p  = A / (B + EPS);
  float r  = C / (D + EPS);
  float cb = 2.f * p * r / (p + r + EPS);
  float loss = 1.f - cb;
  for (int off = 16; off > 0; off >>= 1) loss += __shfl_down(loss, off);
  if (n == 0) out[0] = loss * (1.f / NIMG);
}

// ---------------------------------------------------------------------------
extern "C" void kernel_launch(void* const* d_in, const int* in_sizes, int n_in,
                              void* d_out, int out_size, void* d_ws, size_t ws_size,
                              hipStream_t stream) {
  (void)in_sizes; (void)n_in; (void)out_size; (void)ws_size;
  const float* P = (const float*)d_in[0];   // pred   (N,1,H,W) f32
  const float* G = (const float*)d_in[1];   // target (N,1,H,W) f32
  float* sums    = (float*)d_ws;            // NIMG*4 accumulators
  float* out     = (float*)d_out;           // scalar f32

  cbdice_zero_ws<<<1, 128, 0, stream>>>(sums, NIMG * 4);

  dim3 grid(W / T, H / T, NIMG);            // 24 x 24 x 32
  cbdice_tile_kernel<<<grid, NTHREADS, 0, stream>>>(P, G, sums);

  cbdice_finalize<<<1, 32, 0, stream>>>(sums, out);
}